// MDNLoss_45165876085017
// MI455X (gfx1250) — compile-verified
//
#include <hip/hip_runtime.h>

typedef __attribute__((ext_vector_type(2))) float v2f;
typedef __attribute__((ext_vector_type(8))) float v8f;

#define NEGINF (-1e30f)

// ---------------------------------------------------------------------------
// Kernel 1 (prep): per (b,t) row,
//   apre[row][2k+0] = exp(-lv[k])            (A operand, even K')
//   apre[row][2k+1] = -2*mu[k]*exp(-lv[k])   (A operand, odd  K')
//   bias[row]       = sum_k ( mu^2*exp(-lv) + lv )
// All transcendental work lives here: O(B*txtT*K) = 163k elements, so the
// O(B*txtT*melT) WMMA kernel below never touches expf.
// ---------------------------------------------------------------------------
__global__ void mdn_prep_kernel(const float* __restrict__ mu_logvar,
                                float* __restrict__ apre,
                                float* __restrict__ bias,
                                int rows, int nmel) {
  int idx = blockIdx.x * blockDim.x + threadIdx.x;
  if (idx >= rows) return;
  const float* row = mu_logvar + (size_t)idx * (2 * nmel);
  float* ap = apre + (size_t)idx * (2 * nmel);
  float s = 0.f;
  for (int k = 0; k < nmel; ++k) {
    float mu = row[k];
    float lv = row[nmel + k];
    float w  = expf(-lv);
    ap[2 * k]     = w;
    ap[2 * k + 1] = -2.0f * mu * w;
    s = fmaf(mu * mu, w, s + lv);
  }
  bias[idx] = s;
}

// ---------------------------------------------------------------------------
// Kernel 2: log_prob_matrix via V_WMMA_F32_16X16X4_F32, precomputed A.
// One wave32 per 16x16 (txt x mel) tile.  Contraction length 2*nmel:
//   even K' = 2k : A = exp(-lv[t,k])        B = x[m,k]^2
//   odd  K' = 2k+1: A = -2*mu[t,k]*exp(-lv) B = x[m,k]
// A 16x4 f32 layout (ISA 7.12.2): lane group g (lane>>4) supplies K'=base+2g+j
// in VGPR j for row M = lane&15; B mirrors it for column N = lane&15.
// C/D layout: VGPR r -> row (r + 8g), col lane&15.
// Inner loop: b64 load (A pair) + b32 load (x) + 1 mul + 1 wmma, with pure
// pointer-bump addressing (no per-iteration 64-bit multiplies).
// ---------------------------------------------------------------------------
__global__ void mdn_logprob_wmma_pre(const float* __restrict__ apre,
                                     const float* __restrict__ melspec,
                                     const float* __restrict__ bias,
                                     float* __restrict__ lpm,
                                     int txtT, int melT, int nmel) {
  const int lane = threadIdx.x;      // 0..31, wave32
  const int g    = lane >> 4;        // half-wave group (0/1)
  const int r    = lane & 15;
  const int b    = blockIdx.z;
  const int t0   = blockIdx.y << 4;
  const int m0   = blockIdx.x << 4;

  // A stream: this lane supplies K' = 4s + 2g + {0,1}  ->  float offset 4s + 2g
  const float* ap = apre + ((size_t)b * txtT + (size_t)(t0 + r)) * (size_t)(2 * nmel)
                         + (size_t)(2 * g);
  // B stream: this lane supplies x[m0+r, k] with k = 2s + g  ->  row stride 2*melT
  const float* mp = melspec + (size_t)b * nmel * melT
                            + (size_t)g * melT + (size_t)(m0 + r);
  const size_t melStride = 2 * (size_t)melT;

  v8f acc = {0.f, 0.f, 0.f, 0.f, 0.f, 0.f, 0.f, 0.f};

  const int steps = nmel >> 1;       // 40 WMMA iterations (K'=160, 4 per step)
  for (int s = 0; s < steps; ++s) {
    const v2f a = *(const v2f*)ap;   // {w, -2*mu*w}, 8B aligned
    ap += 4;
    const float x = *mp;
    mp += melStride;
    v2f bb;
    bb.x = x * x;                    // K' even
    bb.y = x;                        // K' odd
    acc = __builtin_amdgcn_wmma_f32_16x16x4_f32(
        /*neg_a=*/false, a, /*neg_b=*/false, bb,
        /*c_mod=*/(short)0, acc, /*reuse_a=*/false, /*reuse_b=*/false);
  }

  const float scale = -0.5f / (float)nmel;      // -1/(2*nmel)
  const int   m     = m0 + r;
#pragma unroll
  for (int rr = 0; rr < 8; ++rr) {
    const int t = t0 + rr + (g << 3);
    const float bt = bias[b * txtT + t];
    lpm[((size_t)b * txtT + t) * melT + m] = (acc[rr] + bt) * scale;
  }
}

// ---------------------------------------------------------------------------
// Kernel 2b (fallback if workspace too small for apre): fused expf variant.
// ---------------------------------------------------------------------------
__global__ void mdn_logprob_wmma_fused(const float* __restrict__ mu_logvar,
                                       const float* __restrict__ melspec,
                                       const float* __restrict__ bias,
                                       float* __restrict__ lpm,
                                       int txtT, int melT, int nmel) {
  const int lane = threadIdx.x;
  const int g    = lane >> 4;
  const int r    = lane & 15;
  const int b    = blockIdx.z;
  const int t0   = blockIdx.y << 4;
  const int m0   = blockIdx.x << 4;

  const float* mlv = mu_logvar + ((size_t)b * txtT + (size_t)(t0 + r)) * (size_t)(2 * nmel);
  const float* pmu = mlv + g;               // k = 2s + g
  const float* plv = mlv + nmel + g;
  const float* mp  = melspec + (size_t)b * nmel * melT
                             + (size_t)g * melT + (size_t)(m0 + r);
  const size_t melStride = 2 * (size_t)melT;

  v8f acc = {0.f, 0.f, 0.f, 0.f, 0.f, 0.f, 0.f, 0.f};

  const int steps = nmel >> 1;
  for (int s = 0; s < steps; ++s) {
    const float mu = *pmu;  pmu += 2;
    const float lv = *plv;  plv += 2;
    const float w  = expf(-lv);
    v2f a;
    a.x = w;
    a.y = -2.0f * mu * w;
    const float x = *mp;    mp += melStride;
    v2f bb;
    bb.x = x * x;
    bb.y = x;
    acc = __builtin_amdgcn_wmma_f32_16x16x4_f32(
        false, a, false, bb, (short)0, acc, false, false);
  }

  const float scale = -0.5f / (float)nmel;
  const int   m     = m0 + r;
#pragma unroll
  for (int rr = 0; rr < 8; ++rr) {
    const int t = t0 + rr + (g << 3);
    const float bt = bias[b * txtT + t];
    lpm[((size_t)b * txtT + t) * melT + m] = (acc[rr] + bt) * scale;
  }
}

// ---------------------------------------------------------------------------
// Kernel 1b (fallback prep): bias only.
// ---------------------------------------------------------------------------
__global__ void mdn_bias_kernel(const float* __restrict__ mu_logvar,
                                float* __restrict__ bias,
                                int rows, int nmel) {
  int idx = blockIdx.x * blockDim.x + threadIdx.x;
  if (idx >= rows) return;
  const float* row = mu_logvar + (size_t)idx * (2 * nmel);
  float s = 0.f;
  for (int k = 0; k < nmel; ++k) {
    float mu = row[k];
    float lv = row[nmel + k];
    s = fmaf(mu * mu, expf(-lv), s + lv);
  }
  bias[idx] = s;
}

// ---------------------------------------------------------------------------
// Kernel 3: forward log-alpha scan.  One block per batch, one thread per txt
// position.  melT-1 sequential steps; shift-by-one through LDS.
// ---------------------------------------------------------------------------
__global__ void mdn_scan_kernel(const float* __restrict__ lpm,
                                const int* __restrict__ text_len,
                                const int* __restrict__ mel_len,
                                float* __restrict__ alpha_norm,
                                int txtT, int melT) {
  extern __shared__ float sh[];      // txtT + 1 floats
  const int b  = blockIdx.x;
  const int t  = threadIdx.x;
  const int TL = text_len[b];
  const int ML = mel_len[b];
  const float* lp = lpm + (size_t)b * txtT * melT;

  float prev = (t == 0) ? lp[0] : NEGINF;    // init row: alpha[:,0]=lp[0,0]
  float captured = NEGINF;
  if (ML == 1 && t == TL - 1) captured = prev;

  for (int m = 1; m < melT; ++m) {
    sh[t + 1] = prev;
    if (t == 0) sh[0] = NEGINF;
    __syncthreads();
    const float shifted = sh[t];
    const float mx = fmaxf(prev, shifted);
    const float mn = fminf(prev, shifted);
    const float cur = mx + log1pf(expf(mn - mx)) + lp[(size_t)t * melT + m];
    __syncthreads();
    prev = cur;
    if (m == ML - 1 && t == TL - 1) captured = cur;
  }
  if (t == TL - 1) alpha_norm[b] = captured / (float)ML;
}

// ---------------------------------------------------------------------------
// Kernel 4: mdn_loss = -mean_b(alpha_norm[b])
// ---------------------------------------------------------------------------
__global__ void mdn_reduce_kernel(const float* __restrict__ alpha_norm,
                                  float* __restrict__ out, int B) {
  if (threadIdx.x == 0) {
    float s = 0.f;
    for (int b = 0; b < B; ++b) s += alpha_norm[b];
    out[0] = -s / (float)B;
  }
}

// ---------------------------------------------------------------------------
extern "C" void kernel_launch(void* const* d_in, const int* in_sizes, int n_in,
                              void* d_out, int out_size, void* d_ws, size_t ws_size,
                              hipStream_t stream) {
  const float* mu_logvar    = (const float*)d_in[0];
  const float* melspec      = (const float*)d_in[1];
  const int*   text_lengths = (const int*)d_in[2];
  const int*   mel_lengths  = (const int*)d_in[3];

  const int nmel = 80;
  const int B    = in_sizes[2];                       // 16
  const int txtT = in_sizes[0] / (B * 2 * nmel);      // 128
  const int melT = in_sizes[1] / (B * nmel);          // 512

  float* out = (float*)d_out;
  float* lpm = out + 1;                               // log_prob_matrix [B,txtT,melT]

  const int rows = B * txtT;
  // Workspace layout: bias[rows] | alpha_norm[B] | apre[rows * 2*nmel]
  float* bias       = (float*)d_ws;
  float* alpha_norm = bias + rows;
  float* apre       = alpha_norm + B;                 // 8B-aligned: (rows+B)*4 % 8 == 0
  const size_t need = ((size_t)rows + B + (size_t)rows * 2 * nmel) * sizeof(float);

  dim3 grid(melT / 16, txtT / 16, B);                 // (32, 8, 16) tiles

  if (ws_size >= need) {
    mdn_prep_kernel<<<(rows + 255) / 256, 256, 0, stream>>>(mu_logvar, apre, bias,
                                                            rows, nmel);
    mdn_logprob_wmma_pre<<<grid, 32, 0, stream>>>(apre, melspec, bias, lpm,
                                                  txtT, melT, nmel);
  } else {
    mdn_bias_kernel<<<(rows + 255) / 256, 256, 0, stream>>>(mu_logvar, bias,
                                                            rows, nmel);
    mdn_logprob_wmma_fused<<<grid, 32, 0, stream>>>(mu_logvar, melspec, bias, lpm,
                                                    txtT, melT, nmel);
  }

  mdn_scan_kernel<<<B, txtT, (txtT + 1) * sizeof(float), stream>>>(
      lpm, text_lengths, mel_lengths, alpha_norm, txtT, melT);

  mdn_reduce_kernel<<<1, 32, 0, stream>>>(alpha_norm, out, B);
}